// EdgeConstructor_33655363731998
// MI455X (gfx1250) — compile-verified
//
#include <hip/hip_runtime.h>
#include <hip/hip_bf16.h>
#include <math.h>

// EdgeConstructor for MI455X (gfx1250, wave32).
// out[b][i][j] = { dR, mass }:
//   dR   = sqrt(max(deta^2 + wrap(dphi)^2, 1e-12))
//   mass = sqrt(max(s_i + s_j + 2*G_ij, 1e-12))
// where G = P * diag(1,-1,-1,-1) * P^T  (P rows = (E,px,py,pz)),
// computed per 16x16 tile with V_WMMA_F32_16X16X4_F32 (K=4 == 4-vector).

typedef float v2f __attribute__((ext_vector_type(2)));
typedef float v8f __attribute__((ext_vector_type(8)));

#define N_NODES 256
#define PI_F      3.14159265358979323846f
#define TWO_PI_F  6.28318530717958647693f
#define INV_2PI_F 0.15915494309189533577f

__global__ __launch_bounds__(256)
void edge_constructor_kernel(const float* __restrict__ x,
                             float* __restrict__ out) {
  const int b = blockIdx.x;

  __shared__ float4 sVec[N_NODES];  // (E, px, py, pz)
  __shared__ float4 sAux[N_NODES];  // (eta, phi, s = E^2-|p|^2, 0)

  const int tid = threadIdx.x;

  // ---- Stage per-node quantities into LDS (coalesced B128 load) ----
  {
    const float4 in = reinterpret_cast<const float4*>(x)[(size_t)b * N_NODES + tid];
    const float pt = in.x, eta = in.y, phi = in.z, E = in.w;
    float sphi, cphi;
    __sincosf(phi, &sphi, &cphi);
    const float px = pt * cphi;
    const float py = pt * sphi;
    const float pz = pt * sinhf(eta);
    const float s  = E * E - px * px - py * py - pz * pz;
    sVec[tid] = make_float4(E, px, py, pz);
    sAux[tid] = make_float4(eta, phi, s, 0.0f);
  }
  __syncthreads();

  const int wave = tid >> 5;        // 8 waves per block
  const int lane = tid & 31;
  const int half = lane >> 4;       // 0: lanes 0-15 (K=0,1), 1: lanes 16-31 (K=2,3)
  const int l16  = lane & 15;       // row (A) / col (B) index within tile

  float2* __restrict__ out2 = reinterpret_cast<float2*>(out);

  // Each wave handles 32 of the 256 16x16 tiles of the 256x256 pair plane.
  const int tEnd = wave * 32 + 32;
  for (int t = wave * 32; t < tEnd; ++t) {
    const int ti = t >> 4;
    const int tj = t & 15;
    const int i0 = ti << 4;
    const int j0 = tj << 4;

    // ---- A fragment: row i = i0 + l16, metric-signed (E, -px, -py, -pz) ----
    // 32-bit A 16x4 layout: lanes 0-15 hold K=0 (V0), K=1 (V1);
    //                       lanes 16-31 hold K=2 (V0), K=3 (V1).
    const float4 va = sVec[i0 + l16];
    v2f afrag;
    if (half == 0) { afrag[0] = va.x;  afrag[1] = -va.y; }   // ( E, -px)
    else           { afrag[0] = -va.z; afrag[1] = -va.w; }   // (-py, -pz)

    // ---- B fragment: col j = j0 + l16, plain (E, px, py, pz) ----
    const int  j  = j0 + l16;
    const float4 vb = sVec[j];
    v2f bfrag;
    if (half == 0) { bfrag[0] = vb.x; bfrag[1] = vb.y; }
    else           { bfrag[0] = vb.z; bfrag[1] = vb.w; }

    // ---- G tile: one WMMA, K = 4 (the full 4-vector) ----
    v8f acc = {};
    acc = __builtin_amdgcn_wmma_f32_16x16x4_f32(
        /*neg_a=*/false, afrag, /*neg_b=*/false, bfrag,
        /*c_mod=*/(short)0, acc, /*reuse_a=*/false, /*reuse_b=*/false);

    // Column aux is constant across the 8 accumulator rows for this lane.
    const float4 cj = sAux[j];  // (eta_j, phi_j, s_j, -)

    // C/D layout: VGPR v, lane L -> M = v + 8*(L/16), N = L%16.
    #pragma unroll
    for (int v = 0; v < 8; ++v) {
      const int i = i0 + v + (half << 3);
      const float4 ri = sAux[i];  // (eta_i, phi_i, s_i, -)

      const float deta = ri.x - cj.x;
      float dphi = ri.y - cj.y + PI_F;
      dphi = dphi - TWO_PI_F * floorf(dphi * INV_2PI_F);  // jnp.mod(., 2pi)
      dphi -= PI_F;

      const float dR   = sqrtf(fmaxf(deta * deta + dphi * dphi, 1e-12f));
      const float m2   = ri.z + cj.z + 2.0f * acc[v];
      const float mass = sqrtf(fmaxf(m2, 1e-12f));

      // Coalesced: 16 lanes x 8B = 128B contiguous per half-wave.
      const size_t idx = (((size_t)b * N_NODES + i) * N_NODES + j);
      out2[idx] = make_float2(dR, mass);
    }
  }
}

extern "C" void kernel_launch(void* const* d_in, const int* in_sizes, int n_in,
                              void* d_out, int out_size, void* d_ws, size_t ws_size,
                              hipStream_t stream) {
  const float* x = reinterpret_cast<const float*>(d_in[0]);
  float* out = reinterpret_cast<float*>(d_out);
  // x is [B, 256, 4] fp32.
  const int B = in_sizes[0] / (N_NODES * 4);
  edge_constructor_kernel<<<B, 256, 0, stream>>>(x, out);
}